// RNN_71425306132974
// MI455X (gfx1250) — compile-verified
//
#include <hip/hip_runtime.h>
#include <math.h>
#include <stdint.h>

#define VOCABN 32000
#define EMBN   256
#define HIDN   512
#define BATCHN 32
#define SEQN   256
#define NWG_RNN 32   // 32 WGs x 16 hidden units = 512
#define KC     32    // K-chunk staged through LDS in out_gemm
#define LDSPAD 2     // row padding (floats) -> 34-float stride, conflict-free

typedef float __attribute__((ext_vector_type(2))) v2f;
typedef float __attribute__((ext_vector_type(4))) v4f;
typedef float __attribute__((ext_vector_type(8))) v8f;

__device__ __forceinline__ v8f wmma_f32(v2f a, v2f b, v8f c) {
  // D = A(16x4 f32) * B(4x16 f32) + C(16x16 f32), exact fp32 matrix pipe
  return __builtin_amdgcn_wmma_f32_16x16x4_f32(false, a, false, b, (short)0, c,
                                               false, false);
}

// Per-lane async copy of 16 bytes from global gptr to the LDS byte address lds_byte_addr, counted by ASYNCcnt
__device__ __forceinline__ void async_g2l_b128(unsigned lds_byte_addr,
                                               const float* gptr) {
  unsigned long long ga = (unsigned long long)(uintptr_t)gptr;
  asm volatile("global_load_async_to_lds_b128 %0, %1, off"
               :: "v"(lds_byte_addr), "v"(ga)
               : "memory");
}

// ---------------------------------------------------------------------------
// Persistent GRU recurrence: 32 WGs x 64 threads (2 waves). Each WG owns 16
// hidden units; each wave owns a 16-row batch tile. Device-wide barrier per
// timestep via monotonic atomic counter (all 32 tiny WGs trivially co-resident).
// ---------------------------------------------------------------------------
__global__ void rnn_persistent(const int* __restrict__ ids,
                               const float* __restrict__ emb,
                               const float* __restrict__ Wih,
                               const float* __restrict__ Whh,
                               const float* __restrict__ bih,
                               const float* __restrict__ bhh,
                               float* __restrict__ hbuf,   // 2 * B*H ping-pong
                               float* __restrict__ Hmat,   // [B*T, H]
                               unsigned* __restrict__ bar)
{
  const int lane = threadIdx.x & 31;
  const int wave = threadIdx.x >> 5;      // 0..1
  const int m0   = wave * 16;             // batch tile base
  const int l15  = lane & 15;
  const int kh   = (lane >> 4) << 1;      // K half select: 0 or 2
  const int hi8  = (lane >> 4) << 3;      // D row offset: 0 or 8
  const int j0   = blockIdx.x * 16;       // hidden-unit tile base
  const int unit = j0 + l15;              // B-frag column (hidden unit)
  const int brow = m0 + l15;              // A-frag row (batch)

  const float bir = bih[unit],            bhr = bhh[unit];
  const float biz = bih[HIDN + unit],     bhz = bhh[HIDN + unit];
  const float bin = bih[2 * HIDN + unit], bhn = bhh[2 * HIDN + unit];

  const float* w_ir = Wih + (size_t)(0 * HIDN + unit) * EMBN;
  const float* w_iz = Wih + (size_t)(1 * HIDN + unit) * EMBN;
  const float* w_in = Wih + (size_t)(2 * HIDN + unit) * EMBN;
  const float* w_hr = Whh + (size_t)(0 * HIDN + unit) * HIDN;
  const float* w_hz = Whh + (size_t)(1 * HIDN + unit) * HIDN;
  const float* w_hn = Whh + (size_t)(2 * HIDN + unit) * HIDN;

  for (int t = 0; t < SEQN; ++t) {
    const float* hprev = hbuf + (size_t)(t & 1) * (BATCHN * HIDN);
    float*       hnext = hbuf + (size_t)((t + 1) & 1) * (BATCHN * HIDN);

    const int tok = (t == 0) ? 0 : ids[brow * SEQN + (t - 1)];
    const float* xrow = emb + (size_t)tok * EMBN;
    const float* hrow = hprev + (size_t)brow * HIDN;

    v8f aR = {}, aZ = {}, aN = {};   // gi accumulators (r,z,n)
    v8f hR = {}, hZ = {}, hN = {};   // gh accumulators (r,z,n)

    #pragma unroll 4
    for (int k = 0; k < EMBN; k += 4) {
      v2f a  = *(const v2f*)(xrow + k + kh);
      v2f b0 = *(const v2f*)(w_ir + k + kh);
      v2f b1 = *(const v2f*)(w_iz + k + kh);
      v2f b2 = *(const v2f*)(w_in + k + kh);
      aR = wmma_f32(a, b0, aR);
      aZ = wmma_f32(a, b1, aZ);
      aN = wmma_f32(a, b2, aN);
    }
    #pragma unroll 4
    for (int k = 0; k < HIDN; k += 4) {
      v2f a  = *(const v2f*)(hrow + k + kh);
      v2f b0 = *(const v2f*)(w_hr + k + kh);
      v2f b1 = *(const v2f*)(w_hz + k + kh);
      v2f b2 = *(const v2f*)(w_hn + k + kh);
      hR = wmma_f32(a, b0, hR);
      hZ = wmma_f32(a, b1, hZ);
      hN = wmma_f32(a, b2, hN);
    }

    #pragma unroll
    for (int i = 0; i < 8; ++i) {
      const int bb = m0 + hi8 + i;   // batch row of this D element
      const float gr = (aR[i] + bir) + (hR[i] + bhr);
      const float gz = (aZ[i] + biz) + (hZ[i] + bhz);
      const float r  = 1.0f / (1.0f + expf(-gr));
      const float z  = 1.0f / (1.0f + expf(-gz));
      const float n  = tanhf((aN[i] + bin) + r * (hN[i] + bhn));
      const float hp = hprev[(size_t)bb * HIDN + unit];
      const float hv = (1.0f - z) * n + z * hp;
      hnext[(size_t)bb * HIDN + unit] = hv;
      Hmat[((size_t)bb * SEQN + t) * HIDN + unit] = hv;
    }

    // ---- device-wide barrier (release writes, acquire peers' writes) ----
    __syncthreads();
    if (threadIdx.x == 0) {
      __hip_atomic_fetch_add(bar, 1u, __ATOMIC_ACQ_REL, __HIP_MEMORY_SCOPE_AGENT);
      const unsigned target = (unsigned)(t + 1) * NWG_RNN;
      while (__hip_atomic_load(bar, __ATOMIC_ACQUIRE, __HIP_MEMORY_SCOPE_AGENT)
             < target)
        __builtin_amdgcn_s_sleep(1);
    }
    __syncthreads();
  }
}

// ---------------------------------------------------------------------------
// Logits GEMM: [8192,512] x [512,32000] + bias, exact f32 WMMA.
// WG = 256 threads = 8 waves as 2(M) x 4(N); wave computes 32x32 (4 tiles).
// W_out K-slabs are staged to LDS with double-buffered async-to-LDS copies
// (ASYNCcnt), shared by all 8 waves; A streams from L2; output write-bound.
// ---------------------------------------------------------------------------
__global__ void out_gemm(const float* __restrict__ Hmat,
                         const float* __restrict__ Wout,
                         const float* __restrict__ bout,
                         float* __restrict__ out)
{
  __shared__ float lb[2][128][KC + LDSPAD];   // 2 x 128 rows x 32 floats (+pad)

  const int lane = threadIdx.x & 31;
  const int wave = threadIdx.x >> 5;   // 0..7
  const int l15  = lane & 15;
  const int kh   = (lane >> 4) << 1;
  const int hi8  = (lane >> 4) << 3;

  const int nblk = blockIdx.x * 128;                  // WG col base
  const int m0   = blockIdx.y * 64 + (wave >> 2) * 32;
  const int n0   = nblk + (wave & 3) * 32;
  const int nl0  = (wave & 3) * 32 + l15;             // local B rows in LDS
  const int nl1  = nl0 + 16;

  const float* a0p = Hmat + (size_t)(m0 + l15) * HIDN;
  const float* a1p = Hmat + (size_t)(m0 + 16 + l15) * HIDN;

  // async stage of one K-chunk of W_out: 128 rows x 32 floats = 16KB
  auto issue = [&](int c, int buf) {
    const int kbase = c * KC;
    #pragma unroll
    for (int q = 0; q < 4; ++q) {
      const int slot = (int)threadIdx.x + 256 * q;    // 0..1023
      const int row  = slot >> 3;                     // 0..127
      const int kq   = (slot & 7) << 2;               // 0,4,...,28
      const float* gp = Wout + (size_t)(nblk + row) * HIDN + kbase + kq;
      const unsigned lds = (unsigned)(uintptr_t)&lb[buf][row][kq];
      async_g2l_b128(lds, gp);
    }
  };

  v8f c00 = {}, c01 = {}, c10 = {}, c11 = {};

  const int NCH = HIDN / KC;   // 16
  issue(0, 0);
  for (int c = 0; c < NCH; ++c) {
    const int buf = c & 1;
    if (c + 1 < NCH) {
      issue(c + 1, buf ^ 1);
      asm volatile("s_wait_asynccnt 0x4" ::: "memory");
    } else {
      asm volatile("s_wait_asynccnt 0x0" ::: "memory");
    }
    __syncthreads();

    const int kbase = c * KC;
    #pragma unroll
    for (int kk = 0; kk < KC; kk += 4) {
      v2f a0 = *(const v2f*)(a0p + kbase + kk + kh);
      v2f a1 = *(const v2f*)(a1p + kbase + kk + kh);
      v2f b0 = *(const v2f*)&lb[buf][nl0][kk + kh];
      v2f b1 = *(const v2f*)&lb[buf][nl1][kk + kh];
      c00 = wmma_f32(a0, b0, c00);
      c01 = wmma_f32(a0, b1, c01);
      c10 = wmma_f32(a1, b0, c10);
      c11 = wmma_f32(a1, b1, c11);
    }
    __syncthreads();   // buffer `buf` may be overwritten at iteration c+1
  }

  const float bias0 = bout[n0 + l15];
  const float bias1 = bout[n0 + 16 + l15];
  #pragma unroll
  for (int i = 0; i < 8; ++i) {
    const int r0 = m0 + hi8 + i;
    const int r1 = m0 + 16 + hi8 + i;
    out[(size_t)r0 * VOCABN + (n0 + l15)]      = c00[i] + bias0;
    out[(size_t)r0 * VOCABN + (n0 + 16 + l15)] = c01[i] + bias1;
    out[(size_t)r1 * VOCABN + (n0 + l15)]      = c10[i] + bias0;
    out[(size_t)r1 * VOCABN + (n0 + 16 + l15)] = c11[i] + bias1;
  }
}

// ---------------------------------------------------------------------------
// Per-row online logsumexp (deterministic, no atomics): alpha = max + log(sum)
// ---------------------------------------------------------------------------
__global__ void row_lse(const float* __restrict__ out, float* __restrict__ alpha)
{
  __shared__ float sm[256];
  __shared__ float ss[256];
  const int row = blockIdx.x;
  const float* p = out + (size_t)row * VOCABN;

  float m = -INFINITY, s = 0.0f;
  for (int i = threadIdx.x; i < VOCABN; i += 256) {
    const float x = p[i];
    if (x > m) { s = s * expf(m - x) + 1.0f; m = x; }
    else       { s += expf(x - m); }
  }
  sm[threadIdx.x] = m; ss[threadIdx.x] = s;
  __syncthreads();
  for (int off = 128; off > 0; off >>= 1) {
    if (threadIdx.x < off) {
      const float m1 = sm[threadIdx.x],        s1 = ss[threadIdx.x];
      const float m2 = sm[threadIdx.x + off],  s2 = ss[threadIdx.x + off];
      const float mm = fmaxf(m1, m2);
      sm[threadIdx.x] = mm;
      ss[threadIdx.x] = s1 * expf(m1 - mm) + s2 * expf(m2 - mm);
    }
    __syncthreads();
  }
  if (threadIdx.x == 0) alpha[row] = sm[0] + logf(ss[0]);
}

// ---------------------------------------------------------------------------
// In-place: logprob = logit - alpha[row]; float4 vectorized (VOCAB % 4 == 0)
// ---------------------------------------------------------------------------
__global__ void sub_alpha(float* __restrict__ out, const float* __restrict__ alpha)
{
  const size_t i = ((size_t)blockIdx.x * blockDim.x + threadIdx.x) * 4;
  const int row = (int)(i / VOCABN);
  const float a = alpha[row];
  v4f v = *(v4f*)(out + i);
  v.x -= a; v.y -= a; v.z -= a; v.w -= a;
  *(v4f*)(out + i) = v;
}

extern "C" void kernel_launch(void* const* d_in, const int* in_sizes, int n_in,
                              void* d_out, int out_size, void* d_ws, size_t ws_size,
                              hipStream_t stream)
{
  (void)in_sizes; (void)n_in; (void)out_size; (void)ws_size;

  const int*   ids  = (const int*)  d_in[0];
  const float* emb  = (const float*)d_in[1];
  const float* Wih  = (const float*)d_in[2];
  const float* Whh  = (const float*)d_in[3];
  const float* bih  = (const float*)d_in[4];
  const float* bhh  = (const float*)d_in[5];
  const float* Wout = (const float*)d_in[6];
  const float* bout = (const float*)d_in[7];
  float* out = (float*)d_out;

  // Workspace layout: [0,256) barrier | h ping-pong (128KB) | H (16.8MB) | alpha
  char* ws = (char*)d_ws;
  const size_t HBUF_OFF = 256;
  const size_t HBUF_BYTES = (size_t)2 * BATCHN * HIDN * sizeof(float);
  const size_t HMAT_OFF = HBUF_OFF + HBUF_BYTES;                 // 131328
  const size_t HMAT_BYTES = (size_t)BATCHN * SEQN * HIDN * sizeof(float);
  const size_t ALPHA_OFF = HMAT_OFF + HMAT_BYTES;

  unsigned* bar  = (unsigned*)ws;
  float* hbuf    = (float*)(ws + HBUF_OFF);
  float* Hmat    = (float*)(ws + HMAT_OFF);
  float* alpha   = (float*)(ws + ALPHA_OFF);

  // Zero barrier + h0 every call (graph-capturable memset node)
  hipMemsetAsync(ws, 0, HBUF_OFF + HBUF_BYTES, stream);

  rnn_persistent<<<NWG_RNN, 64, 0, stream>>>(ids, emb, Wih, Whh, bih, bhh,
                                             hbuf, Hmat, bar);

  out_gemm<<<dim3(VOCABN / 128, (BATCHN * SEQN) / 64), 256, 0, stream>>>(
      Hmat, Wout, bout, out);

  row_lse<<<BATCHN * SEQN, 256, 0, stream>>>(out, alpha);

  const int n_vec4 = (int)(((size_t)BATCHN * SEQN * VOCABN) / 4 / 256); // 256000
  sub_alpha<<<n_vec4, 256, 0, stream>>>(out, alpha);
}